// SelfAttention_10531259810686
// MI455X (gfx1250) — compile-verified
//
#include <hip/hip_runtime.h>
#include <cmath>

typedef float v2f __attribute__((ext_vector_type(2)));
typedef float v8f __attribute__((ext_vector_type(8)));
typedef unsigned int u32x4 __attribute__((ext_vector_type(4)));
typedef int i32x4 __attribute__((ext_vector_type(4)));
typedef int i32x8 __attribute__((ext_vector_type(8)));

#define EPSY (1.0f / 137.0f)

__device__ __forceinline__ v8f wmma_f32(v2f a, v2f b, v8f c) {
  // D = A(16x4) * B(4x16) + C(16x16), fp32, wave32
  return __builtin_amdgcn_wmma_f32_16x16x4_f32(false, a, false, b, (short)0, c,
                                               false, false);
}

__device__ __forceinline__ void lds_fence() {
  asm volatile("s_wait_dscnt 0" ::: "memory");
}

// Per-lane async global->LDS copy (ASYNCcnt). 16B granule, 16B-aligned LDS dst.
__device__ __forceinline__ void async_load_b128(void* lds, const void* g) {
  asm volatile("global_load_async_to_lds_b128 %0, %1, off" ::"v"(
                   (unsigned int)(uintptr_t)lds),
               "v"(g)
               : "memory");
}
__device__ __forceinline__ void async_wait0() {
  asm volatile("s_wait_asynccnt 0x0" ::: "memory");
}

// ---------------------------------------------------------------------------
// TDM: DMA a 2D tile (tile_d0 x tile_d1 fp32 elements, row stride
// stride_d0 elements) from global to LDS, optional LDS row padding.
// Descriptor bitfields per CDNA5 ISA D# group0/group1 tables.
// ---------------------------------------------------------------------------
__device__ __forceinline__ void tdm_load_2d(unsigned int lds_off,
                                            const void* gaddr,
                                            unsigned int tile_d0,
                                            unsigned int tile_d1,
                                            unsigned long long stride_d0,
                                            unsigned int pad_interval_enc,
                                            unsigned int pad_amount_enc,
                                            bool pad_en) {
  unsigned long long ga = (unsigned long long)(uintptr_t)gaddr;
  u32x4 g0;
  g0[0] = 1u;                  // count=1 (valid), user mode, no gather
  g0[1] = lds_off;             // lds_addr
  g0[2] = (unsigned int)ga;    // global_addr[31:0]
  g0[3] = (unsigned int)((ga >> 32) & 0x1FFFFFFu) | (2u << 30);  // type=2
  unsigned int w0 = (2u << 16);  // workgroup_mask=0, data_size=4B
  if (pad_en)
    w0 |= (1u << 20) | (pad_interval_enc << 22) | (pad_amount_enc << 25);
  i32x8 g1;
  g1[0] = (int)w0;
  g1[1] = (int)((tile_d0 & 0xFFFFu) << 16);  // tensor_dim0[15:0] @ [63:48]
  g1[2] = (int)(((tile_d0 >> 16) & 0xFFFFu) |
                ((tile_d1 & 0xFFFFu) << 16));  // dim0 hi16 | tensor_dim1 lo16
  g1[3] = (int)(((tile_d1 >> 16) & 0xFFFFu) |
                ((tile_d0 & 0xFFFFu) << 16));  // dim1 hi16 | tile_dim0
  g1[4] = (int)(tile_d1 & 0xFFFFu);            // tile_dim1 | tile_dim2=0
  g1[5] = (int)(unsigned int)stride_d0;        // tensor_dim0_stride lo32
  g1[6] = (int)((unsigned int)(stride_d0 >> 32) & 0xFFFFu);  // stride hi16
  g1[7] = 0;
  i32x4 z4 = {0, 0, 0, 0};
#if __clang_major__ >= 23
  i32x8 z8 = {0, 0, 0, 0, 0, 0, 0, 0};
  __builtin_amdgcn_tensor_load_to_lds(g0, g1, z4, z4, z8, 0);
#else
  __builtin_amdgcn_tensor_load_to_lds(g0, g1, z4, z4, 0);
#endif
}

// ---------------------------------------------------------------------------
// Column squared-norms of W[K][N]:  k2[n] = sum_k W[k][n]^2
// ---------------------------------------------------------------------------
__global__ void colnorm_kernel(const float* __restrict__ w,
                               float* __restrict__ out, int K, int N) {
  int n = blockIdx.x * blockDim.x + threadIdx.x;
  if (n >= N) return;
  float s = 0.f;
  for (int k = 0; k < K; ++k) {
    float v = w[(size_t)k * N + n];
    s += v * v;
  }
  out[n] = s;
}

// ---------------------------------------------------------------------------
// Row squared-norms of A[rows][cols]: out[r] = sum_c A[r][c]^2  (block per row)
// ---------------------------------------------------------------------------
__global__ __launch_bounds__(256) void rownorm_kernel(
    const float* __restrict__ a, float* __restrict__ out, int cols) {
  __shared__ float red[8];
  int row = blockIdx.x;
  const float* p = a + (size_t)row * cols;
  float s = 0.f;
  for (int c = threadIdx.x; c < cols; c += blockDim.x) {
    float v = p[c];
    s += v * v;
  }
  s += __shfl_xor(s, 16, 32);
  s += __shfl_xor(s, 8, 32);
  s += __shfl_xor(s, 4, 32);
  s += __shfl_xor(s, 2, 32);
  s += __shfl_xor(s, 1, 32);
  int lane = threadIdx.x & 31, wave = threadIdx.x >> 5;
  if (lane == 0) red[wave] = s;
  __syncthreads();
  if (threadIdx.x == 0) {
    float t = 0.f;
    for (int i = 0; i < 8; ++i) t += red[i];
    out[row] = t;
  }
}

// ---------------------------------------------------------------------------
// Yat-dense GEMM:  out = yat(A[M,K] @ W[K,N])
//   mode 0: scatter into permuted q/k/v buffers [B,64,T,16]
//   mode 1: write out[M,N] directly
// Block tile 64x128, 8 waves (2x4), wave tile 32x32 (2x2 WMMA subtiles),
// K-step 16, double-buffered LDS fed by the Tensor Data Mover (TENSORcnt).
// ---------------------------------------------------------------------------
__global__ __launch_bounds__(256) void yat_gemm_kernel(
    const float* __restrict__ A, const float* __restrict__ W,
    const float* __restrict__ x2, const float* __restrict__ k2,
    const float* __restrict__ bias, const float* __restrict__ alpha,
    float* __restrict__ outd, float* __restrict__ qp, float* __restrict__ kp,
    float* __restrict__ vp, int M, int N, int K, int mode) {
  __shared__ __align__(16) float As[2][64][18];   // 16 + 2 pad (TDM pad)
  __shared__ __align__(16) float Bs[2][16][128];

  const int tid = threadIdx.x;
  const int lane = tid & 31;
  const int wave = tid >> 5;
  const int wm = wave >> 2;  // 0..1
  const int wn = wave & 3;   // 0..3
  const int bn = blockIdx.x * 128;
  const int bm = blockIdx.y * 64;

  v8f acc[2][2];
#pragma unroll
  for (int i = 0; i < 2; ++i)
#pragma unroll
    for (int j = 0; j < 2; ++j)
#pragma unroll
      for (int e = 0; e < 8; ++e) acc[i][j][e] = 0.f;

  auto issue_stage = [&](int st, int k0) {
    if (wave == 0) {
      // A tile: 64 rows x 16 elems, row stride K, LDS pad 16->18 dwords
      tdm_load_2d((unsigned int)(uintptr_t)&As[st][0][0],
                  A + (size_t)bm * K + k0, 16u, 64u, (unsigned long long)K,
                  3u /*16 dw*/, 1u /*+2 dw*/, true);
      // B tile: 16 rows x 128 elems, row stride N, contiguous LDS rows
      tdm_load_2d((unsigned int)(uintptr_t)&Bs[st][0][0],
                  W + (size_t)k0 * N + bn, 128u, 16u, (unsigned long long)N, 0u,
                  0u, false);
    }
  };

  issue_stage(0, 0);
  if (wave == 0) __builtin_amdgcn_s_wait_tensorcnt(0);
  __syncthreads();

  const int steps = K >> 4;
  for (int kt = 0; kt < steps; ++kt) {
    int cur = kt & 1;
    if (kt + 1 < steps) issue_stage(cur ^ 1, (kt + 1) << 4);
#pragma unroll
    for (int c = 0; c < 4; ++c) {
      int kk = c * 4 + ((lane >> 4) << 1);  // even
      v2f afr[2], bfr[2];
#pragma unroll
      for (int sm = 0; sm < 2; ++sm) {
        const float* p = &As[cur][wm * 32 + sm * 16 + (lane & 15)][kk];
        float2 t = *(const float2*)p;
        afr[sm].x = t.x;
        afr[sm].y = t.y;
      }
#pragma unroll
      for (int sn = 0; sn < 2; ++sn) {
        int n = wn * 32 + sn * 16 + (lane & 15);
        bfr[sn].x = Bs[cur][kk][n];
        bfr[sn].y = Bs[cur][kk + 1][n];
      }
#pragma unroll
      for (int sm = 0; sm < 2; ++sm)
#pragma unroll
        for (int sn = 0; sn < 2; ++sn)
          acc[sm][sn] = wmma_f32(afr[sm], bfr[sn], acc[sm][sn]);
    }
    if (wave == 0) __builtin_amdgcn_s_wait_tensorcnt(0);
    __syncthreads();
  }

  // Yat epilogue
  float al = alpha[0];
  float sc = __powf(sqrtf((float)N) / log1pf((float)N), al);
#pragma unroll
  for (int sm = 0; sm < 2; ++sm) {
#pragma unroll
    for (int sn = 0; sn < 2; ++sn) {
      int mloc = wm * 32 + sm * 16 + ((lane >> 4) << 3);
      int n = bn + wn * 32 + sn * 16 + (lane & 15);
      float k2n = k2[n];
      float bsn = bias[n];
#pragma unroll
      for (int v = 0; v < 8; ++v) {
        int m = bm + mloc + v;
        float y = acc[sm][sn][v];
        float val = (y * y) / (x2[m] + k2n - 2.f * y + EPSY) * sc + bsn;
        if (mode == 0) {
          int s = n >> 10, r = n & 1023, h = r >> 6, g = r & 63;
          int b = m >> 10, t = m & 1023;
          float* dst = (s == 0) ? qp : ((s == 1) ? kp : vp);
          dst[(((size_t)b * 64 + g) * 1024 + t) * 16 + h] = val;
        } else {
          outd[(size_t)m * N + n] = val;
        }
      }
    }
  }
}

// ---------------------------------------------------------------------------
// Flash-style yat attention over the permuted layout:
//   per (b,g): Q,K,V = [T,16];  S = (QK^T)^2/(q2+k2-2QK^T+eps)*isc, causal,
//   online softmax, O = P V.
// WG: 8 waves, wave = one 16-query block; K/V super-blocks (128 keys) double-
// buffered in LDS via per-lane async global->LDS copies (ASYNCcnt).
// ---------------------------------------------------------------------------
__global__ __launch_bounds__(256) void yat_attn_kernel(
    const float* __restrict__ qp, const float* __restrict__ kp,
    const float* __restrict__ vp, const float* __restrict__ alpha_attn,
    float* __restrict__ out) {
  __shared__ __align__(16) float Ks[2][128][20];  // pad 20 -> 16B-aligned rows
  __shared__ __align__(16) float Vs[2][128][20];
  __shared__ __align__(16) float Ps[8][16][18];
  __shared__ float q2s[8][16];

  const int tid = threadIdx.x;
  const int lane = tid & 31;
  const int wave = tid >> 5;
  const int bg = blockIdx.x;  // b*64 + g
  const int b = bg >> 6;
  const int g = bg & 63;
  const int qc = blockIdx.y;
  const int q0 = qc * 128 + wave * 16;

  // Q fragments (A layout), 4 chunks of K=4
  v2f qf[4];
  float part = 0.f;
#pragma unroll
  for (int c = 0; c < 4; ++c) {
    int m = lane & 15;
    int d = c * 4 + ((lane >> 4) << 1);
    const float* p = qp + ((size_t)bg * 1024 + q0 + m) * 16 + d;
    float2 t = *(const float2*)p;
    qf[c].x = t.x;
    qf[c].y = t.y;
    part += t.x * t.x + t.y * t.y;
  }
  // q2 per row m = lane&15 (lanes l and l+16 hold complementary dims)
  float q2full = part + __shfl_xor(part, 16, 32);
  if (lane < 16) q2s[wave][lane] = q2full;

  auto prefetch_kv = [&](int buf, int sb) {
    int row = tid >> 1, hb = (tid & 1) << 3;
    const float* kpp = kp + ((size_t)bg * 1024 + sb * 128 + row) * 16 + hb;
    const float* vpp = vp + ((size_t)bg * 1024 + sb * 128 + row) * 16 + hb;
    async_load_b128(&Ks[buf][row][hb], kpp);
    async_load_b128(&Ks[buf][row][hb + 4], kpp + 4);
    async_load_b128(&Vs[buf][row][hb], vpp);
    async_load_b128(&Vs[buf][row][hb + 4], vpp + 4);
  };

  const int nsb = qc + 1;  // causal: key super-blocks 0..qc
  prefetch_kv(0, 0);
  async_wait0();
  __syncthreads();  // K/V buf0 + q2s visible

  float q2v[8];
#pragma unroll
  for (int v = 0; v < 8; ++v) q2v[v] = q2s[wave][v + ((lane >> 4) << 3)];

  float isc = __powf(64.0f / log1pf(64.0f), alpha_attn[0]);

  float mrow[8], lrow[8];
  v8f oacc;
#pragma unroll
  for (int v = 0; v < 8; ++v) {
    mrow[v] = -1e30f;
    lrow[v] = 0.f;
    oacc[v] = 0.f;
  }

  for (int sb = 0; sb < nsb; ++sb) {
    int cur = sb & 1;
    if (sb + 1 < nsb) prefetch_kv(cur ^ 1, sb + 1);

    for (int lb = 0; lb < 8; ++lb) {
      int kb = sb * 128 + lb * 16;
      if (kb > q0) break;  // wave-uniform causal skip
      int n = lane & 15;
      int jr = lb * 16 + n;

      // k2 for this lane's key column
      float k2l = 0.f;
#pragma unroll
      for (int d = 0; d < 16; d += 2) {
        float2 kv = *(const float2*)&Ks[cur][jr][d];
        k2l += kv.x * kv.x + kv.y * kv.y;
      }

      // S = Q K^T (B fragment = K rows as columns)
      v8f s;
#pragma unroll
      for (int e = 0; e < 8; ++e) s[e] = 0.f;
#pragma unroll
      for (int c = 0; c < 4; ++c) {
        int d = c * 4 + ((lane >> 4) << 1);
        v2f bf;
        float2 t = *(const float2*)&Ks[cur][jr][d];
        bf.x = t.x;
        bf.y = t.y;
        s = wmma_f32(qf[c], bf, s);
      }

      // yat logits + causal mask + online softmax (row = M, col = key)
#pragma unroll
      for (int v = 0; v < 8; ++v) {
        int Mr = v + ((lane >> 4) << 3);
        int qi = q0 + Mr;
        int kj = kb + n;
        float dot = s[v];
        float logit = (dot * dot) / (q2v[v] + k2l - 2.f * dot + EPSY) * isc;
        bool masked = kj > qi;
        float xv = masked ? -3.0e38f : logit;
        float r = xv;
        r = fmaxf(r, __shfl_xor(r, 1, 32));
        r = fmaxf(r, __shfl_xor(r, 2, 32));
        r = fmaxf(r, __shfl_xor(r, 4, 32));
        r = fmaxf(r, __shfl_xor(r, 8, 32));
        float mnew = fmaxf(mrow[v], r);
        float f = __expf(mrow[v] - mnew);
        float p = masked ? 0.f : __expf(logit - mnew);
        float psum = p;
        psum += __shfl_xor(psum, 1, 32);
        psum += __shfl_xor(psum, 2, 32);
        psum += __shfl_xor(psum, 4, 32);
        psum += __shfl_xor(psum, 8, 32);
        lrow[v] = lrow[v] * f + psum;
        mrow[v] = mnew;
        oacc[v] = oacc[v] * f;
        Ps[wave][Mr][n] = p;  // stage P (C layout -> LDS)
      }
      lds_fence();  // wave-local: P stores before A-layout reads

      // O += P V
#pragma unroll
      for (int c = 0; c < 4; ++c) {
        int d = c * 4 + ((lane >> 4) << 1);
        v2f ap;
        float2 t = *(const float2*)&Ps[wave][lane & 15][d];
        ap.x = t.x;
        ap.y = t.y;
        v2f bv;
        bv.x = Vs[cur][lb * 16 + d][n];
        bv.y = Vs[cur][lb * 16 + d + 1][n];
        oacc = wmma_f32(ap, bv, oacc);
      }
    }

    async_wait0();   // next super-block landed
    __syncthreads(); // everyone done with buffer `cur`
  }

  // normalize + write out channel c = g*16 + feature
#pragma unroll
  for (int v = 0; v < 8; ++v) {
    int Mr = v + ((lane >> 4) << 3);
    int t = q0 + Mr;
    float val = oacc[v] / lrow[v];
    out[((size_t)b * 1024 + t) * 1024 + g * 16 + (lane & 15)] = val;
  }
}

// ---------------------------------------------------------------------------
extern "C" void kernel_launch(void* const* d_in, const int* in_sizes, int n_in,
                              void* d_out, int out_size, void* d_ws,
                              size_t ws_size, hipStream_t stream) {
  (void)in_sizes;
  (void)n_in;
  (void)out_size;
  (void)ws_size;
  const float* x = (const float*)d_in[0];
  // d_in[1] = mask (bool causal) -- implemented analytically
  const float* w_qkv = (const float*)d_in[2];
  const float* b_qkv = (const float*)d_in[3];
  const float* alpha_qkv = (const float*)d_in[4];
  const float* w_proj = (const float*)d_in[5];
  const float* b_proj = (const float*)d_in[6];
  const float* alpha_proj = (const float*)d_in[7];
  const float* alpha_attn = (const float*)d_in[8];

  const int B = 2, T = 1024, C = 1024;
  const int M = B * T, N3 = 3 * C;

  float* ws = (float*)d_ws;
  float* qp = ws;
  ws += (size_t)B * 64 * T * 16;  // [B,64,T,16]
  float* kp = ws;
  ws += (size_t)B * 64 * T * 16;
  float* vp = ws;
  ws += (size_t)B * 64 * T * 16;
  float* ao = ws;
  ws += (size_t)M * C;  // attention output [B,T,C]
  float* x2 = ws;
  ws += M;
  float* xo2 = ws;
  ws += M;
  float* k2q = ws;
  ws += N3;
  float* k2p = ws;
  ws += C;

  colnorm_kernel<<<N3 / 256, 256, 0, stream>>>(w_qkv, k2q, C, N3);
  colnorm_kernel<<<C / 256, 256, 0, stream>>>(w_proj, k2p, C, C);
  rownorm_kernel<<<M, 256, 0, stream>>>(x, x2, C);

  yat_gemm_kernel<<<dim3(N3 / 128, M / 64), 256, 0, stream>>>(
      x, w_qkv, x2, k2q, b_qkv, alpha_qkv, nullptr, qp, kp, vp, M, N3, C, 0);

  yat_attn_kernel<<<dim3(B * 64, T / 128), 256, 0, stream>>>(qp, kp, vp,
                                                             alpha_attn, ao);

  rownorm_kernel<<<M, 256, 0, stream>>>(ao, xo2, C);

  yat_gemm_kernel<<<dim3(C / 128, M / 64), 256, 0, stream>>>(
      ao, w_proj, xo2, k2p, b_proj, alpha_proj, (float*)d_out, nullptr, nullptr,
      nullptr, M, C, C, 1);
}